// SAGPooling_23751169147227
// MI455X (gfx1250) — compile-verified
//
#include <hip/hip_runtime.h>
#include <hip/hip_bf16.h>
#include <stdint.h>

// ---------------- problem constants (match reference) ----------------
#define B_   32
#define N_   2048
#define D_   256
#define DEG_ 32
#define E_   (B_ * N_ * DEG_)     // 2,097,152
#define DE_  32
#define K_   1024
#define NUM_NODES (B_ * N_)       // 65,536
#define NPOOL     (B_ * K_)       // 32,768

// ---------------- d_out layout (float elements, in return order) -----
#define OFF_X    ((size_t)0)                          // NPOOL*D_      = 8,388,608
#define OFF_EI   ((size_t)(NPOOL) * D_)               // 2*E_          @ 8,388,608
#define OFF_EA   (OFF_EI + (size_t)2 * E_)            // E_*DE_        @ 12,582,912
#define OFF_ET   (OFF_EA + (size_t)E_ * DE_)          // E_            @ 79,691,776
#define OFF_NT   (OFF_ET + (size_t)E_)                // NPOOL         @ 81,788,928
#define OFF_B    (OFF_NT + (size_t)NPOOL)             // NPOOL         @ 81,821,696
#define OFF_PERM (OFF_B + (size_t)NPOOL)              // NPOOL         @ 81,854,464
#define OFF_SP   (OFF_PERM + (size_t)NPOOL)           // NPOOL         @ 81,887,232

// ---------------- CDNA5 async global->LDS helpers ---------------------
// LDS byte offset: low 32 bits of a generic pointer to __shared__ are the
// in-LDS offset (generic LDS addr = {aperture, offset32}).
__device__ __forceinline__ uint32_t lds_addr_of(const void* p) {
  return (uint32_t)(uintptr_t)p;
}
__device__ __forceinline__ void async_load_b128(uint32_t lds, const void* gaddr) {
  asm volatile("global_load_async_to_lds_b128 %0, %1, off"
               :: "v"(lds), "v"(gaddr) : "memory");
}
__device__ __forceinline__ void wait_async_le1() {
  asm volatile("s_wait_asynccnt 1" ::: "memory");
}
__device__ __forceinline__ void wait_async_le0() {
  asm volatile("s_wait_asynccnt 0" ::: "memory");
}

// =====================================================================
// Kernel 1: per-graph tanh + full bitonic sort (desc, tie: low index
// first, matching lax.top_k) of 2048 (val,idx) pairs in LDS.
// One 1024-thread (32 x wave32) workgroup per graph. Emits:
//   perm (ws + out), score_t[perm], node_type[perm], batch[perm],
//   and the FUSED new_id[] map: after the sort each node of graph g
//   appears exactly once in si[0..2047], so thread t writes
//   new_id[g*N+si[t]] = g*K+t (kept) and new_id[g*N+si[t+1024]] = -1
//   (dropped) — race-free, no init/scatter kernels needed.
// =====================================================================
__global__ __launch_bounds__(1024) void k_topk(
    const float* __restrict__ score, const int* __restrict__ node_type,
    float* __restrict__ out, int* __restrict__ perm_ws,
    int* __restrict__ nid) {
  __shared__ float sv[N_];
  __shared__ int   si[N_];
  const int g = blockIdx.x;
  const int t = threadIdx.x;

#pragma unroll
  for (int s = 0; s < 2; ++s) {
    int i = t + s * 1024;
    sv[i] = tanhf(score[g * N_ + i]);
    si[i] = i;
  }
  __syncthreads();

  for (int k = 2; k <= N_; k <<= 1) {
    for (int j = k >> 1; j > 0; j >>= 1) {
#pragma unroll
      for (int s = 0; s < 2; ++s) {
        int i = t + s * 1024;
        int ixj = i ^ j;
        if (ixj > i) {
          float av = sv[i], bv = sv[ixj];
          int   ai = si[i], bi = si[ixj];
          // b_first: element at ixj should precede element at i in the
          // final order (descending value, ascending index on ties).
          bool b_first = (bv > av) || ((bv == av) && (bi < ai));
          bool do_swap = ((i & k) == 0) ? b_first : !b_first;
          if (do_swap) {
            sv[i] = bv; sv[ixj] = av;
            si[i] = bi; si[ixj] = ai;
          }
        }
      }
      __syncthreads();
    }
  }

  // entries 0..K_-1 are the ordered top-K; K_..N_-1 are dropped nodes.
  int   li = si[t];
  float v  = sv[t];
  int   pg = g * N_ + li;      // global node id of kept node, rank t
  int   r  = g * K_ + t;       // pooled row id
  perm_ws[r]        = pg;
  out[OFF_PERM + r] = (float)pg;
  out[OFF_SP   + r] = v;
  out[OFF_NT   + r] = (float)node_type[pg];
  out[OFF_B    + r] = (float)g;
  // fused new_id construction (each node written exactly once)
  nid[pg]                     = r;    // kept  -> new compact id
  nid[g * N_ + si[t + 1024]]  = -1;   // dropped
}

// =====================================================================
// Kernel 2: x_out[r] = x[perm[r]] * score_t[perm[r]]
// Gather streamed through LDS with CDNA5 async global->LDS b128 loads,
// 2-deep double buffer (s_wait_asynccnt 1/0). 64 lanes per row (float4),
// 4 rows per 256-thread tile, XITER tiles per block.
// =====================================================================
#define XITER 16
__global__ __launch_bounds__(256) void k_xout(
    const float* __restrict__ x, const int* __restrict__ perm,
    const float* __restrict__ sp, float* __restrict__ xout) {
  __shared__ float4 buf[2][256];
  const int tid    = threadIdx.x;
  const int lane_c = tid & 63;   // float4 column within row
  const int rsub   = tid >> 6;   // row within 4-row tile
  const int base_row = blockIdx.x * (4 * XITER);

  uint32_t lds0 = lds_addr_of(&buf[0][tid]);
  uint32_t lds1 = lds_addr_of(&buf[1][tid]);

  // prologue: prefetch tile 0
  {
    int r = base_row + rsub;
    const float4* g = (const float4*)x + (size_t)perm[r] * (D_ / 4) + lane_c;
    async_load_b128(lds0, (const void*)g);
  }
  for (int it = 0; it < XITER; ++it) {
    int r = base_row + it * 4 + rsub;
    if (it + 1 < XITER) {
      int rn = base_row + (it + 1) * 4 + rsub;
      const float4* g = (const float4*)x + (size_t)perm[rn] * (D_ / 4) + lane_c;
      async_load_b128(((it + 1) & 1) ? lds1 : lds0, (const void*)g);
      wait_async_le1();   // oldest (current tile) load is complete
    } else {
      wait_async_le0();
    }
    float  s = sp[r];
    float4 v = buf[it & 1][tid];
    v.x *= s; v.y *= s; v.z *= s; v.w *= s;
    ((float4*)xout)[(size_t)r * (D_ / 4) + lane_c] = v;
  }
}

// =====================================================================
// Kernel 3: edge filtering: mask, remapped edge_index (-1 if dropped),
// edge_type (-1 if dropped); mask byte to workspace for kernel 4.
// =====================================================================
__global__ __launch_bounds__(256) void k_edges(
    const int* __restrict__ src, const int* __restrict__ dst,
    const int* __restrict__ etype, const int* __restrict__ nid,
    float* __restrict__ out, unsigned char* __restrict__ mask) {
  int e = blockIdx.x * blockDim.x + threadIdx.x;
  if (e >= E_) return;
  int nr = nid[src[e]];
  int nc = nid[dst[e]];
  bool m = (nr >= 0) && (nc >= 0);
  out[OFF_EI + e]              = m ? (float)nr : -1.0f;
  out[OFF_EI + (size_t)E_ + e] = m ? (float)nc : -1.0f;
  out[OFF_ET + e]              = m ? (float)etype[e] : -1.0f;
  mask[e] = m ? 1 : 0;
}

// =====================================================================
// Kernel 4: edge_attr masked copy (the big stream: 268 MB out).
// Per-lane EXEC-predicated b128 loads: ~75% of edges are dropped, so the
// hardware skips fetching their cachelines entirely (~200 MB HBM saved).
// =====================================================================
__global__ __launch_bounds__(256) void k_eattr(
    const float4* __restrict__ ea, const unsigned char* __restrict__ mask,
    float4* __restrict__ out) {
  size_t slot = (size_t)blockIdx.x * 256 + threadIdx.x;  // E_*8 slots of 16B
  if (slot >= (size_t)E_ * (DE_ / 4)) return;
  float4 v = make_float4(0.0f, 0.0f, 0.0f, 0.0f);
  if (mask[slot >> 3]) v = ea[slot];
  out[slot] = v;
}

// =====================================================================
extern "C" void kernel_launch(void* const* d_in, const int* in_sizes, int n_in,
                              void* d_out, int out_size, void* d_ws, size_t ws_size,
                              hipStream_t stream) {
  const float* x     = (const float*)d_in[0];
  const float* score = (const float*)d_in[1];
  const int*   ei    = (const int*)d_in[2];   // [2, E_]: src = ei, dst = ei + E_
  const float* ea    = (const float*)d_in[3];
  const int*   etype = (const int*)d_in[4];
  const int*   ntype = (const int*)d_in[5];
  // d_in[6] (batch) unused: batch[perm] == graph id by construction.

  float* out = (float*)d_out;

  // workspace layout (fully rewritten every call -> replay-safe)
  int*           ws_perm = (int*)d_ws;                                      // 32768 i32
  int*           ws_nid  = (int*)((char*)d_ws + 131072);                    // 65536 i32
  unsigned char* ws_mask = (unsigned char*)((char*)d_ws + 131072 + 262144); // E_ u8

  k_topk <<<B_, 1024, 0, stream>>>(score, ntype, out, ws_perm, ws_nid);
  k_xout <<<NPOOL / (4 * XITER), 256, 0, stream>>>(x, ws_perm,
                                                   out + OFF_SP, out + OFF_X);
  k_edges<<<E_ / 256, 256, 0, stream>>>(ei, ei + E_, etype, ws_nid, out, ws_mask);
  k_eattr<<<(E_ * (DE_ / 4)) / 256, 256, 0, stream>>>((const float4*)ea, ws_mask,
                                                      (float4*)(out + OFF_EA));
}